// RRNNforGRUCell_23862838297165
// MI455X (gfx1250) — compile-verified
//
#include <hip/hip_runtime.h>

#define H_DIM 1024
#define B_DIM 2048
#define NC    4

typedef float v2f __attribute__((ext_vector_type(2)));
typedef float v4f __attribute__((ext_vector_type(4)));
typedef float v8f __attribute__((ext_vector_type(8)));

// ---------------------------------------------------------------------------
// f32 WMMA GEMM:  C[z][M=2048][N=1024] (+)= A[M][K=1024] @ W[z][K][N]
// Block tile 128x128x32, 256 threads = 8 wave32s (2M x 4N wave grid), each
// wave computes 64x32 = 4x2 accumulators via V_WMMA_F32_16X16X4_F32.
// Register-staged double-buffered LDS pipeline, one barrier per K-step.
// ---------------------------------------------------------------------------
#define BM 128
#define BN 128
#define BK 32
#define LDA (BK + 4)   // 36 floats -> 144B row stride (16B aligned, conflict-free frags)
#define LDB (BN + 4)   // 132 floats -> 528B row stride (16B aligned)
#define ASZ (BM * LDA) // 4608 floats
#define BSZ (BK * LDB) // 4224 floats

template <bool ACC>
__global__ __launch_bounds__(256) void gemm_f32_wmma(
    const float* __restrict__ A, const float* __restrict__ W,
    float* __restrict__ C, long wStride, long cStride) {
  __shared__ __align__(16) float smem[2 * (ASZ + BSZ)];

  const int tid  = threadIdx.x;
  const int bm0  = blockIdx.x * BM;
  const int bn0  = blockIdx.y * BN;
  const float* Wk = W + (long)blockIdx.z * wStride;
  float*       Ck = C + (long)blockIdx.z * cStride;

  const int wid   = tid >> 5;
  const int lane  = tid & 31;
  const int waveM = (wid >> 2) * 64;   // 0 or 64
  const int waveN = (wid & 3) * 32;    // 0,32,64,96
  const int lrow  = lane & 15;
  const int lgrp  = lane >> 4;         // 0: K,K+1 ; 1: K+2,K+3
  const int kOff  = lgrp * 2;

  // Per-thread tile-load coordinates (fixed across K-steps).
  int aRow[4], aKq[4], bKr[4], bNq[4];
#pragma unroll
  for (int i = 0; i < 4; ++i) {
    int linear = tid + i * 256;
    aRow[i] = linear >> 3;            // 0..127
    aKq[i]  = (linear & 7) << 2;      // 0..28
    bKr[i]  = linear >> 5;            // 0..31
    bNq[i]  = (linear & 31) << 2;     // 0..124
  }

  v8f acc[4][2] = {};
  v4f pa[4], pb[4];

  // ---- prologue: tile 0 -> LDS buffer 0 ----
#pragma unroll
  for (int i = 0; i < 4; ++i)
    pa[i] = *(const v4f*)(A + (long)(bm0 + aRow[i]) * H_DIM + aKq[i]);
#pragma unroll
  for (int i = 0; i < 4; ++i)
    pb[i] = *(const v4f*)(Wk + (long)bKr[i] * H_DIM + bn0 + bNq[i]);
  {
    float* As = smem;
    float* Bs = smem + ASZ;
#pragma unroll
    for (int i = 0; i < 4; ++i) *(v4f*)(As + aRow[i] * LDA + aKq[i]) = pa[i];
#pragma unroll
    for (int i = 0; i < 4; ++i) *(v4f*)(Bs + bKr[i] * LDB + bNq[i]) = pb[i];
  }
  __syncthreads();

  int cur = 0;
  for (int k0 = 0; k0 < H_DIM; k0 += BK) {
    const bool hasNext = (k0 + BK) < H_DIM;
    // ---- issue next tile's global loads (latency overlapped with WMMAs) ----
    if (hasNext) {
      int kn = k0 + BK;
#pragma unroll
      for (int i = 0; i < 4; ++i)
        pa[i] = *(const v4f*)(A + (long)(bm0 + aRow[i]) * H_DIM + kn + aKq[i]);
#pragma unroll
      for (int i = 0; i < 4; ++i)
        pb[i] = *(const v4f*)(Wk + (long)(kn + bKr[i]) * H_DIM + bn0 + bNq[i]);
    }

    // ---- compute from the current LDS buffer (cur) ----
    const float* As = smem + cur * (ASZ + BSZ);
    const float* Bs = As + ASZ;
#pragma unroll
    for (int ks = 0; ks < BK; ks += 4) {
      v2f af[4];
#pragma unroll
      for (int mt = 0; mt < 4; ++mt) {
        int m = waveM + mt * 16 + lrow;
        af[mt] = *(const v2f*)(As + m * LDA + ks + kOff);
      }
      v2f bf[2];
#pragma unroll
      for (int nt = 0; nt < 2; ++nt) {
        int n = waveN + nt * 16 + lrow;
        bf[nt].x = Bs[(ks + kOff) * LDB + n];
        bf[nt].y = Bs[(ks + kOff + 1) * LDB + n];
      }
#pragma unroll
      for (int mt = 0; mt < 4; ++mt)
#pragma unroll
        for (int nt = 0; nt < 2; ++nt)
          acc[mt][nt] = __builtin_amdgcn_wmma_f32_16x16x4_f32(
              false, af[mt], false, bf[nt], (short)0, acc[mt][nt], false, false);
    }

    // ---- stage next tile into the other LDS buffer; single barrier ----
    if (hasNext) {
      float* Asw = smem + (cur ^ 1) * (ASZ + BSZ);
      float* Bsw = Asw + ASZ;
#pragma unroll
      for (int i = 0; i < 4; ++i) *(v4f*)(Asw + aRow[i] * LDA + aKq[i]) = pa[i];
#pragma unroll
      for (int i = 0; i < 4; ++i) *(v4f*)(Bsw + bKr[i] * LDB + bNq[i]) = pb[i];
      __syncthreads();
      cur ^= 1;
    }
  }

  // ---- epilogue: C VGPR v holds M = v + 8*lgrp, N = lrow ----
#pragma unroll
  for (int mt = 0; mt < 4; ++mt) {
#pragma unroll
    for (int nt = 0; nt < 2; ++nt) {
      int col = bn0 + waveN + nt * 16 + lrow;
#pragma unroll
      for (int v = 0; v < 8; ++v) {
        int row = bm0 + waveM + mt * 16 + v + lgrp * 8;
        long idx = (long)row * H_DIM + col;
        float val = acc[mt][nt][v];
        if (ACC) val += Ck[idx];
        Ck[idx] = val;
      }
    }
  }
}

// ---------------------------------------------------------------------------
// Elementwise helpers
// ---------------------------------------------------------------------------
__global__ void zero_g_kernel(int* __restrict__ G, int n) {
  int i = blockIdx.x * 256 + threadIdx.x;
  if (i < n) G[i] = 0;
}

__global__ void sigmoid_bias_kernel(float* __restrict__ v,
                                    const float* __restrict__ brow, int n) {
  int i = blockIdx.x * 256 + threadIdx.x;
  if (i < n) {
    float t = v[i] + brow[i & (H_DIM - 1)];
    v[i] = 1.0f / (1.0f + expf(-t));
  }
}

// combine + softmax-mix:  t[k][h] = combine(tA,tC,bias); scores = t.w ;
// p = softmax_k(scores); out[b][h] = sum_k p[k] t[k][h]; G[b][gcol] = argmax
#define MODE_MUL      0
#define MODE_ADDTANH  1
#define MODE_ONEMINUS 2
#define MODE_ADD      3

__global__ __launch_bounds__(256) void mix_kernel(
    const float* __restrict__ tA, const float* __restrict__ tC,
    const float* __restrict__ bias, const float* __restrict__ w,
    float* __restrict__ outv, int* __restrict__ G, int gcol, int mode) {
  __shared__ float tl[NC][H_DIM];
  __shared__ float red[256];
  __shared__ float sc[NC];
  __shared__ float pr[NC];

  const int b   = blockIdx.x;
  const int tid = threadIdx.x;

  for (int k = 0; k < NC; ++k) {
    const float* a = tA + ((long)k * B_DIM + b) * H_DIM;
    const float* c = tC ? tC + ((long)k * B_DIM + b) * H_DIM : nullptr;
    const float* bk = bias + k * H_DIM;
    float p = 0.0f;
    for (int h = tid; h < H_DIM; h += 256) {
      float av = a[h];
      float bv = bk[h];
      float t;
      if (mode == MODE_MUL)           t = av * c[h] + bv;
      else if (mode == MODE_ADDTANH)  t = tanhf(av + c[h] + bv);
      else if (mode == MODE_ONEMINUS) t = 1.0f - (av + bv);
      else                            t = av + c[h] + bv;
      tl[k][h] = t;
      p += t * w[h];
    }
    red[tid] = p;
    __syncthreads();
    for (int s = 128; s > 0; s >>= 1) {
      if (tid < s) red[tid] += red[tid + s];
      __syncthreads();
    }
    if (tid == 0) sc[k] = red[0];
    __syncthreads();
  }

  if (tid == 0) {
    float m = sc[0];
    int gi = 0;
    for (int k = 1; k < NC; ++k)
      if (sc[k] > m) { m = sc[k]; gi = k; }   // first-max like jnp.argmax
    float s = 0.0f;
    for (int k = 0; k < NC; ++k) { pr[k] = expf(sc[k] - m); s += pr[k]; }
    float inv = 1.0f / s;
    for (int k = 0; k < NC; ++k) pr[k] *= inv;
    G[b * 9 + gcol] = gi;
  }
  __syncthreads();

  for (int h = tid; h < H_DIM; h += 256) {
    outv[(long)b * H_DIM + h] =
        pr[0] * tl[0][h] + pr[1] * tl[1][h] + pr[2] * tl[2][h] + pr[3] * tl[3][h];
  }
}

// ---------------------------------------------------------------------------
// Orchestration
// ---------------------------------------------------------------------------
extern "C" void kernel_launch(void* const* d_in, const int* in_sizes, int n_in,
                              void* d_out, int out_size, void* d_ws, size_t ws_size,
                              hipStream_t stream) {
  const float* x  = (const float*)d_in[0];
  const float* hp = (const float*)d_in[1];
  const float* L  = (const float*)d_in[2];
  const float* R  = (const float*)d_in[3];
  const float* bb = (const float*)d_in[4];
  const float* w  = (const float*)d_in[5];

  const long BH = (long)B_DIM * H_DIM;
  const long HH = (long)H_DIM * H_DIM;

  float* hnext = (float*)d_out;
  int*   G     = (int*)d_out + BH;

  float* ws   = (float*)d_ws;
  float* LIN0 = ws;                 // 4*BH
  float* LIN1 = ws + 4 * BH;        // 4*BH
  float* LIN2 = ws + 8 * BH;        // 4*BH
  float* V0   = ws + 12 * BH;       // BH
  float* V1   = ws + 13 * BH;       // BH
  float* V2   = ws + 14 * BH;       // BH

  dim3 blk(256);
  dim3 g1(B_DIM / BM, H_DIM / BN, 1);
  dim3 g4(B_DIM / BM, H_DIM / BN, NC);
  int nvec = (int)BH;

  zero_g_kernel<<<(B_DIM * 9 + 255) / 256, blk, 0, stream>>>(G, B_DIM * 9);

  // gates (note z2 == z1 in the reference, so computed once)
  gemm_f32_wmma<false><<<g1, blk, 0, stream>>>(x,  L,      V0, HH, BH); // x@L0
  gemm_f32_wmma<true ><<<g1, blk, 0, stream>>>(hp, R,      V0, HH, BH); // += h@R0
  gemm_f32_wmma<false><<<g1, blk, 0, stream>>>(x,  L + HH, V1, HH, BH); // x@L1
  gemm_f32_wmma<true ><<<g1, blk, 0, stream>>>(hp, R + HH, V1, HH, BH); // += h@R1
  sigmoid_bias_kernel<<<(nvec + 255) / 256, blk, 0, stream>>>(V0, bb + 0 * H_DIM, nvec); // z1
  sigmoid_bias_kernel<<<(nvec + 255) / 256, blk, 0, stream>>>(V1, bb + 1 * H_DIM, nvec); // r

  // z1R (reused twice), hsL (reused twice)
  gemm_f32_wmma<false><<<g4, blk, 0, stream>>>(V0, R, LIN1, HH, BH);    // z1R
  gemm_f32_wmma<false><<<g4, blk, 0, stream>>>(hp, L, LIN0, HH, BH);    // hsL

  // z2h = mix(hsL * z1R + b), g7 -> V2
  mix_kernel<<<B_DIM, blk, 0, stream>>>(LIN0, LIN1, bb, w, V2, G, 7, MODE_MUL);

  // rh = mix(hsL * rR + b), g3 -> V1 (r consumed by the GEMM first)
  gemm_f32_wmma<false><<<g4, blk, 0, stream>>>(V1, R, LIN2, HH, BH);    // rR
  mix_kernel<<<B_DIM, blk, 0, stream>>>(LIN0, LIN2, bb, w, V1, G, 3, MODE_MUL);

  // one_minus_z1 = mix(1 - (z1R + b)), g5 -> V0
  mix_kernel<<<B_DIM, blk, 0, stream>>>(LIN1, nullptr, bb, w, V0, G, 5, MODE_ONEMINUS);

  // h_tilde = mix(tanh(xsL + rhR + b)), g4 -> V1
  gemm_f32_wmma<false><<<g4, blk, 0, stream>>>(x,  L, LIN0, HH, BH);    // xsL
  gemm_f32_wmma<false><<<g4, blk, 0, stream>>>(V1, R, LIN2, HH, BH);    // rhR
  mix_kernel<<<B_DIM, blk, 0, stream>>>(LIN0, LIN2, bb, w, V1, G, 4, MODE_ADDTANH);

  // zh_tilde = mix(htL * omzR + b), g6 -> V0
  gemm_f32_wmma<false><<<g4, blk, 0, stream>>>(V1, L, LIN0, HH, BH);    // htL
  gemm_f32_wmma<false><<<g4, blk, 0, stream>>>(V0, R, LIN2, HH, BH);    // omzR
  mix_kernel<<<B_DIM, blk, 0, stream>>>(LIN0, LIN2, bb, w, V0, G, 6, MODE_MUL);

  // h_next = mix(zhL + z2hR + b), g8 -> d_out
  gemm_f32_wmma<false><<<g4, blk, 0, stream>>>(V0, L, LIN0, HH, BH);    // zhL
  gemm_f32_wmma<false><<<g4, blk, 0, stream>>>(V2, R, LIN2, HH, BH);    // z2hR
  mix_kernel<<<B_DIM, blk, 0, stream>>>(LIN0, LIN2, bb, w, hnext, G, 8, MODE_ADD);
}